// MEAN_58652073394401
// MI455X (gfx1250) — compile-verified
//
#include <hip/hip_runtime.h>

typedef __attribute__((ext_vector_type(2))) float v2f;
typedef __attribute__((ext_vector_type(8))) float v8f;

#define D       128
#define NREL    16
#define NLAYER  2
#define TM      16          // edges per wave (WMMA M)
#define NWAVE   4           // waves per block
#define BLK_E   (TM*NWAVE)  // edges per block
#define ROWF    (D + 4)     // LDS row stride (pad 4 floats -> conflict-free ds_b64)

// ---------------------------------------------------------------- sort / deg
__global__ void hist_kernel(const int* __restrict__ etype, int* __restrict__ counts, int E) {
    int e = blockIdx.x * blockDim.x + threadIdx.x;
    if (e < E) atomicAdd(&counts[etype[e]], 1);
}

__global__ void scan_kernel(const int* __restrict__ counts, int* __restrict__ offsets,
                            int* __restrict__ cursor) {
    if (threadIdx.x == 0 && blockIdx.x == 0) {
        int acc = 0;
        for (int r = 0; r < NREL; ++r) { offsets[r] = acc; cursor[r] = acc; acc += counts[r]; }
        offsets[NREL] = acc;
    }
}

__global__ void sort_scatter_kernel(const int* __restrict__ etype, const int* __restrict__ dst,
                                    int* __restrict__ cursor, int* __restrict__ eorder,
                                    int* __restrict__ deg, int E) {
    int e = blockIdx.x * blockDim.x + threadIdx.x;
    if (e >= E) return;
    int pos = atomicAdd(&cursor[etype[e]], 1);
    eorder[pos] = e;
    atomicAdd(&deg[dst[e]], 1);
}

// ------------------------------------------------------------- init h gather
__global__ void gather_h_kernel(const float* __restrict__ ent, const int* __restrict__ node_id,
                                float* __restrict__ h, int N) {
    int i = blockIdx.x * blockDim.x + threadIdx.x;   // over N*32 float4
    if (i >= N * (D / 4)) return;
    int node = i >> 5, part = i & 31;
    ((float4*)h)[i] = ((const float4*)(ent + (size_t)node_id[node] * D))[part];
}

// --------------------------------------------------- pack W into WMMA layout
// Wp[((l*NREL + r)*32 + kc)*8 + n][lane][p] = W[l][r][4*kc + 2*(lane>>4) + p][16*n + (lane&15)]
__global__ void pack_w_kernel(const float* __restrict__ W, float* __restrict__ Wp, int total) {
    int t = blockIdx.x * blockDim.x + threadIdx.x;
    if (t >= total) return;
    int p    =  t        & 1;
    int lane = (t >> 1)  & 31;
    int n    = (t >> 6)  & 7;
    int kc   = (t >> 9)  & 31;
    int lr   =  t >> 14;                 // l*NREL + r
    int row  = 4 * kc + 2 * (lane >> 4) + p;
    int col  = 16 * n + (lane & 15);
    Wp[t] = W[((size_t)lr * D + row) * D + col];
}

// ------------------------------------------------------------- WMMA GEMM
// pass 1: z = relu(A@W), accumulate per-channel sum/sumsq
// pass 2: recompute z, msg = relu(z)*scale+shift, atomic scatter-add to agg[dst]
__global__ __launch_bounds__(BLK_E * 2) void rgcn_gemm_kernel(
    const float* __restrict__ h,        // [N,128]
    const float* __restrict__ Wp,       // packed weights for this layer
    const int*   __restrict__ src,
    const int*   __restrict__ dst,
    const int*   __restrict__ eorder,
    const int*   __restrict__ offsets,  // [NREL+1]
    float* __restrict__ gsum, float* __restrict__ gsumsq,
    const float* __restrict__ scale, const float* __restrict__ shift,
    float* __restrict__ agg,
    int pass)
{
    __shared__ __align__(16) float As[NWAVE][TM][ROWF];
    __shared__ int   Dn[NWAVE][TM];
    __shared__ float bsum[D], bsumsq[D];

    const int r        = blockIdx.y;
    const int seg_beg  = offsets[r];
    const int seg_end  = offsets[r + 1];
    const int tile_base = seg_beg + blockIdx.x * BLK_E;
    if (tile_base >= seg_end) return;            // uniform whole-block exit

    const int tid  = threadIdx.x;
    const int w    = tid >> 5;
    const int lane = tid & 31;
    const int lo   = lane & 15;
    const int hi   = lane >> 4;

    bsum[tid] = 0.f; bsumsq[tid] = 0.f;

    // gather 16 source rows into LDS (zero-fill padding rows), remember dst
    const int my_base = tile_base + w * TM;
    for (int j = 0; j < TM; ++j) {
        int e = my_base + j;
        float4 v = make_float4(0.f, 0.f, 0.f, 0.f);
        int dn = 0;
        if (e < seg_end) {
            int es = eorder[e];
            int sn = src[es];
            dn = dst[es];
            v = *(const float4*)(h + (size_t)sn * D + lane * 4);
        }
        *(float4*)&As[w][j][lane * 4] = v;
        if (lane == 0) Dn[w][j] = dn;
    }

    const v8f vzero = {0.f, 0.f, 0.f, 0.f, 0.f, 0.f, 0.f, 0.f};
    v8f acc[8];
#pragma unroll
    for (int n = 0; n < 8; ++n) acc[n] = vzero;

    const float* wp = Wp + (size_t)r * (32 * 8 * 64);
#pragma unroll 4
    for (int kc = 0; kc < 32; ++kc) {
        v2f a = *(const v2f*)&As[w][lo][kc * 4 + 2 * hi];
        const float* bp = wp + kc * (8 * 64) + lane * 2;
#pragma unroll
        for (int n = 0; n < 8; ++n) {
            v2f b = *(const v2f*)(bp + n * 64);
            acc[n] = __builtin_amdgcn_wmma_f32_16x16x4_f32(
                false, a, false, b, (short)0, acc[n], false, false);
        }
    }

    if (pass == 2) {
        const int cnt = seg_end - my_base;       // valid rows in this wave's tile
#pragma unroll
        for (int n = 0; n < 8; ++n) {
            int c = n * 16 + lo;
            float sc = scale[c], sh = shift[c];
#pragma unroll
            for (int j = 0; j < 8; ++j) {
                int row = j + 8 * hi;
                if (row < cnt) {
                    float v = fmaxf(acc[n][j], 0.f) * sc + sh;
                    atomicAdd(agg + (size_t)Dn[w][row] * D + c, v);
                }
            }
        }
    } else {
        // make every wave's bsum/bsumsq zeroing visible before cross-wave LDS atomics
        __syncthreads();
        // per-lane partials (padding rows give relu(0)=0 -> harmless)
#pragma unroll
        for (int n = 0; n < 8; ++n) {
            float s = 0.f, s2 = 0.f;
#pragma unroll
            for (int j = 0; j < 8; ++j) {
                float v = fmaxf(acc[n][j], 0.f);
                s += v; s2 += v * v;
            }
            int c = n * 16 + lo;
            atomicAdd(&bsum[c], s);
            atomicAdd(&bsumsq[c], s2);
        }
        __syncthreads();
        atomicAdd(gsum   + tid, bsum[tid]);
        atomicAdd(gsumsq + tid, bsumsq[tid]);
    }
}

// ------------------------------------------------------------ BN fold
__global__ void stats_kernel(const float* __restrict__ gsum, const float* __restrict__ gsumsq,
                             const float* __restrict__ gamma, const float* __restrict__ beta,
                             float* __restrict__ scale, float* __restrict__ shift, float invE) {
    int t = threadIdx.x;                     // 128 threads
    float mu   = gsum[t] * invE;
    float var  = fmaxf(gsumsq[t] * invE - mu * mu, 0.f);
    float rstd = rsqrtf(var + 1e-5f);
    float sc   = gamma[t] * rstd;
    scale[t] = sc;
    shift[t] = beta[t] - mu * sc;
}

// --------------------------------------------------------- mean + keep-old
__global__ void finalize_kernel(float* __restrict__ h, const float* __restrict__ agg,
                                const int* __restrict__ deg, int N) {
    int i = blockIdx.x * blockDim.x + threadIdx.x;   // over N*128
    if (i >= N * D) return;
    int d = deg[i >> 7];
    if (d > 0) h[i] = agg[i] / (float)d;
}

// ------------------------------------------------------------- output
__global__ void copy_out_kernel(const float4* __restrict__ s, float4* __restrict__ d, int n4) {
    int i = blockIdx.x * blockDim.x + threadIdx.x;
    if (i < n4) d[i] = s[i];
}

__global__ void owner_kernel(const int* __restrict__ node_id, int* __restrict__ owner, int N) {
    int i = blockIdx.x * blockDim.x + threadIdx.x;
    if (i < N) atomicMax(&owner[node_id[i]], i);
}

__global__ void scatter_out_kernel(const float* __restrict__ h, const int* __restrict__ node_id,
                                   const int* __restrict__ owner, float* __restrict__ out, int N) {
    int i = blockIdx.x * blockDim.x + threadIdx.x;   // over N*32 float4
    if (i >= N * (D / 4)) return;
    int node = i >> 5;
    int ent  = node_id[node];
    if (owner[ent] == node)
        ((float4*)out)[(size_t)ent * (D / 4) + (i & 31)] = ((const float4*)h)[i];
}

// ================================================================= launch
extern "C" void kernel_launch(void* const* d_in, const int* in_sizes, int n_in,
                              void* d_out, int out_size, void* d_ws, size_t ws_size,
                              hipStream_t stream) {
    const float* ent     = (const float*)d_in[0];
    const float* W       = (const float*)d_in[1];
    const float* gamma   = (const float*)d_in[2];
    const float* beta    = (const float*)d_in[3];
    const int*   node_id = (const int*)d_in[4];
    const int*   src     = (const int*)d_in[5];
    const int*   dst     = (const int*)d_in[6];
    const int*   etype   = (const int*)d_in[7];

    const int NUM_ENT = in_sizes[0] / D;
    const int N       = in_sizes[4];
    const int E       = in_sizes[5];

    // ---- workspace carve (256B aligned)
    char* p = (char*)d_ws;
    auto carve = [&](size_t bytes) -> void* {
        void* q = (void*)p;
        p += (bytes + 255) & ~(size_t)255;
        return q;
    };
    float* h       = (float*)carve((size_t)N * D * 4);
    float* agg     = (float*)carve((size_t)N * D * 4);
    int*   deg     = (int*)  carve((size_t)N * 4);
    int*   eorder  = (int*)  carve((size_t)E * 4);
    int*   counts  = (int*)  carve(NREL * 4);
    int*   offsets = (int*)  carve((NREL + 1) * 4);
    int*   cursor  = (int*)  carve(NREL * 4);
    float* gsum    = (float*)carve(D * 4);
    float* gsumsq  = (float*)carve(D * 4);
    float* scale   = (float*)carve(D * 4);
    float* shift   = (float*)carve(D * 4);
    const int WP_TOTAL = NLAYER * NREL * 32 * 8 * 64;       // 524288 floats
    float* Wp      = (float*)carve((size_t)WP_TOTAL * 4);
    int*   owner   = (int*)  carve((size_t)NUM_ENT * 4);
    (void)ws_size; (void)n_in; (void)out_size;

    const int T = 256;

    hipMemsetAsync(counts, 0, NREL * 4, stream);
    hipMemsetAsync(deg, 0, (size_t)N * 4, stream);
    hipMemsetAsync(owner, 0xFF, (size_t)NUM_ENT * 4, stream);   // -1

    hist_kernel<<<(E + T - 1) / T, T, 0, stream>>>(etype, counts, E);
    scan_kernel<<<1, 32, 0, stream>>>(counts, offsets, cursor);
    sort_scatter_kernel<<<(E + T - 1) / T, T, 0, stream>>>(etype, dst, cursor, eorder, deg, E);
    gather_h_kernel<<<(N * (D / 4) + T - 1) / T, T, 0, stream>>>(ent, node_id, h, N);
    pack_w_kernel<<<(WP_TOTAL + T - 1) / T, T, 0, stream>>>(W, Wp, WP_TOTAL);

    const dim3 ggrid((E + BLK_E - 1) / BLK_E, NREL);
    const float invE = 1.0f / (float)E;

    for (int l = 0; l < NLAYER; ++l) {
        const float* WpL = Wp + (size_t)l * NREL * 32 * 8 * 64;
        hipMemsetAsync(gsum,   0, D * 4, stream);
        hipMemsetAsync(gsumsq, 0, D * 4, stream);
        hipMemsetAsync(agg,    0, (size_t)N * D * 4, stream);

        rgcn_gemm_kernel<<<ggrid, BLK_E * 2, 0, stream>>>(
            h, WpL, src, dst, eorder, offsets, gsum, gsumsq, scale, shift, agg, 1);
        stats_kernel<<<1, D, 0, stream>>>(gsum, gsumsq, gamma + l * D, beta + l * D,
                                          scale, shift, invE);
        rgcn_gemm_kernel<<<ggrid, BLK_E * 2, 0, stream>>>(
            h, WpL, src, dst, eorder, offsets, gsum, gsumsq, scale, shift, agg, 2);
        finalize_kernel<<<(N * D + T - 1) / T, T, 0, stream>>>(h, agg, deg, N);
    }

    const int n4 = NUM_ENT * (D / 4);
    copy_out_kernel<<<(n4 + T - 1) / T, T, 0, stream>>>((const float4*)ent, (float4*)d_out, n4);
    owner_kernel<<<(N + T - 1) / T, T, 0, stream>>>(node_id, owner, N);
    scatter_out_kernel<<<(N * (D / 4) + T - 1) / T, T, 0, stream>>>(
        h, node_id, owner, (float*)d_out, N);
}